// BERT_CRF_NER_798863917466
// MI455X (gfx1250) — compile-verified
//
#include <hip/hip_runtime.h>
#include <hip/hip_bf16.h>

typedef __attribute__((ext_vector_type(2))) float v2f;
typedef __attribute__((ext_vector_type(8))) float v8f;

#define NEG_INF_F (-10000.0f)
#define Bsz 64
#define Tsz 512
#define Hsz 768
#define Lsz 10
#define LP  16              // labels padded to one WMMA tile width
#define BT  (Bsz * Tsz)     // 32768 tokens
#define WROW 194            // LDS row stride in floats: 194%64==2 -> conflict-free, 8B aligned
#define NCHUNK 64
#define CSTEP  8            // chunk c covers t in [8c, 8c+8) intersect [1,512)

// ---------------------------------------------------------------------------
// Kernel 1: feats[token][l] = sum_h hidden[token][h] * W[l][h] + b[l]
// One wave per 16-token tile; full fp32 via V_WMMA_F32_16X16X4_F32.
// (codegen verified round 1: clause of global_load_b64 + ds_load_2addr_b64 +
//  8x v_wmma per iteration, HBM-roofline bound)
// ---------------------------------------------------------------------------
__global__ __launch_bounds__(256) void feats_wmma_kernel(
    const float* __restrict__ hidden, const float* __restrict__ W,
    const float* __restrict__ bias, float* __restrict__ feats)
{
    __shared__ float Wlds[16 * WROW];

    for (int idx = threadIdx.x; idx < 16 * Hsz; idx += blockDim.x) {
        int n = idx / Hsz, k = idx - n * Hsz;
        Wlds[n * WROW + k] = (n < Lsz) ? W[n * Hsz + k] : 0.0f;
    }
    __syncthreads();

    const int lane = threadIdx.x & 31;
    const int wave = threadIdx.x >> 5;
    const int half = lane >> 4;         // 0: K={k,k+1}, 1: K={k+2,k+3}
    const int r    = lane & 15;         // A: row-in-tile (M), B: column (N)
    const int tile = blockIdx.x * 8 + wave;
    const long rowBase = (long)(tile * 16 + r) * Hsz;

    v8f c = {};
    #pragma unroll 8
    for (int k = 0; k < Hsz; k += 4) {
        const int ka = k + 2 * half;
        v2f a = *(const v2f*)(hidden + rowBase + ka);        // global_load_b64
        v2f bfrag = *(const v2f*)(&Wlds[r * WROW + ka]);     // ds_load_b64
        c = __builtin_amdgcn_wmma_f32_16x16x4_f32(
            false, a, false, bfrag, (short)0, c, false, false);
    }

    const int N = r;
    const float bv = (N < Lsz) ? bias[N] : 0.0f;
    #pragma unroll
    for (int i = 0; i < 8; ++i) {
        const int M = tile * 16 + i + 8 * half;   // C/D layout: VGPR i -> M = i + 8*half
        feats[(long)M * LP + N] = c[i] + bv;
    }
}

// ---------------------------------------------------------------------------
// Kernel 2a: compose each chunk's step operators into a 10x10 log-semiring
// matrix. One wave32 per (batch, chunk) -> 4096 concurrent waves.
// Step operator: A_t[j,i] = trans[j,i] + feat[t,j]
// Compose:       newP[j,i] = feat[t,j] + LSE_k( trans[j,k] + P[k,i] )
// 100 cells striped over 32 lanes (<=4 cells/lane).
// ---------------------------------------------------------------------------
__global__ __launch_bounds__(32) void crf_chunk_kernel(
    const float* __restrict__ trans, const float* __restrict__ feats,
    float* __restrict__ chunkP)
{
    const int bc   = blockIdx.x;
    const int b    = bc >> 6;
    const int c    = bc & 63;
    const int lane = threadIdx.x;

    __shared__ float Tld[Lsz * Lsz];   // transitions, row-major [j][k]
    __shared__ float P[Lsz * Lsz];     // running product, row-major [k][i]

    const float* fb = feats + (long)b * Tsz * LP;
    const int t0 = (c == 0) ? 1 : c * CSTEP;
    const int t1 = c * CSTEP + CSTEP;  // exclusive

    for (int idx = lane; idx < Lsz * Lsz; idx += 32) {
        Tld[idx] = trans[idx];
        // seed P = A_{t0}
        P[idx] = trans[idx] + fb[t0 * LP + idx / Lsz];
    }
    __syncthreads();

    for (int t = t0 + 1; t < t1; ++t) {
        float nv[4];
        int ncell = 0;
        for (int idx = lane; idx < Lsz * Lsz; idx += 32, ++ncell) {
            const int j = idx / Lsz, i = idx - j * Lsz;
            float sc[Lsz];
            sc[0] = Tld[j * Lsz] + P[i];
            float m = sc[0];
            #pragma unroll
            for (int k = 1; k < Lsz; ++k) {
                sc[k] = Tld[j * Lsz + k] + P[k * Lsz + i];
                m = fmaxf(m, sc[k]);
            }
            float s = 0.0f;
            #pragma unroll
            for (int k = 0; k < Lsz; ++k) s += expf(sc[k] - m);
            nv[ncell] = fb[t * LP + j] + m + logf(s);
        }
        __syncthreads();               // all reads of P done
        ncell = 0;
        for (int idx = lane; idx < Lsz * Lsz; idx += 32, ++ncell)
            P[idx] = nv[ncell];
        __syncthreads();
    }

    float* out = chunkP + (long)bc * (Lsz * Lsz);
    for (int idx = lane; idx < Lsz * Lsz; idx += 32) out[idx] = P[idx];
}

// ---------------------------------------------------------------------------
// Kernel 2b: per batch, fold the 64 chunk matrices into alpha (depth 64
// instead of 511), then forward score + gold score.
// ---------------------------------------------------------------------------
__global__ __launch_bounds__(32) void crf_fold_kernel(
    const float* __restrict__ trans, const int* __restrict__ labels,
    const float* __restrict__ feats, const float* __restrict__ chunkP,
    float* __restrict__ scores)
{
    const int b = blockIdx.x;
    const int lane = threadIdx.x;
    __shared__ float alpha[16];

    if (lane < 16) alpha[lane] = (lane == 1 /*START*/) ? 0.0f : NEG_INF_F;
    __syncthreads();

    const float* fb = feats + (long)b * Tsz * LP;

    for (int c = 0; c < NCHUNK; ++c) {
        const float* P = chunkP + ((long)b * NCHUNK + c) * (Lsz * Lsz);
        float nv = 0.0f;
        if (lane < Lsz) {
            float sc[Lsz];
            sc[0] = P[lane * Lsz] + alpha[0];
            float m = sc[0];
            #pragma unroll
            for (int i = 1; i < Lsz; ++i) {
                sc[i] = P[lane * Lsz + i] + alpha[i];
                m = fmaxf(m, sc[i]);
            }
            float s = 0.0f;
            #pragma unroll
            for (int i = 0; i < Lsz; ++i) s += expf(sc[i] - m);
            nv = m + logf(s);
        }
        __syncthreads();
        if (lane < Lsz) alpha[lane] = nv;
        __syncthreads();
    }

    // gold score: lanes stride over t = 1..511
    float g = 0.0f;
    for (int t = 1 + lane; t < Tsz; t += 32) {
        const int cur  = labels[b * Tsz + t];
        const int prev = labels[b * Tsz + t - 1];
        g += trans[cur * Lsz + prev] + fb[t * LP + cur];
    }
    #pragma unroll
    for (int off = 16; off > 0; off >>= 1) g += __shfl_down(g, off, 32);

    if (lane == 0) {
        float m = alpha[0];
        for (int i = 1; i < Lsz; ++i) m = fmaxf(m, alpha[i]);
        float s = 0.0f;
        for (int i = 0; i < Lsz; ++i) s += expf(alpha[i] - m);
        scores[b] = (m + logf(s)) - g;
    }
}

// ---------------------------------------------------------------------------
// Kernel 3: mean over 64 per-batch scores -> scalar
// ---------------------------------------------------------------------------
__global__ __launch_bounds__(32) void reduce_mean_kernel(
    const float* __restrict__ scores, float* __restrict__ out)
{
    const int lane = threadIdx.x;
    float v = scores[lane] + scores[lane + 32];
    #pragma unroll
    for (int off = 16; off > 0; off >>= 1) v += __shfl_down(v, off, 32);
    if (lane == 0) out[0] = v * (1.0f / (float)Bsz);
}

extern "C" void kernel_launch(void* const* d_in, const int* in_sizes, int n_in,
                              void* d_out, int out_size, void* d_ws, size_t ws_size,
                              hipStream_t stream) {
    const float* hidden = (const float*)d_in[0];   // (64,512,768) f32
    const float* W      = (const float*)d_in[1];   // (10,768) f32
    const float* bias   = (const float*)d_in[2];   // (10,) f32
    const float* trans  = (const float*)d_in[3];   // (10,10) f32
    const int*   labels = (const int*)d_in[4];     // (64,512) int

    float* feats  = (float*)d_ws;                                          // 2 MB
    float* chunkP = (float*)((char*)d_ws + (size_t)BT * LP * sizeof(float));      // 1.64 MB
    float* scores = (float*)((char*)d_ws + (size_t)BT * LP * sizeof(float)
                                         + (size_t)Bsz * NCHUNK * Lsz * Lsz * sizeof(float));

    feats_wmma_kernel<<<BT / 16 / 8, 256, 0, stream>>>(hidden, W, bias, feats);
    crf_chunk_kernel<<<Bsz * NCHUNK, 32, 0, stream>>>(trans, feats, chunkP);
    crf_fold_kernel<<<Bsz, 32, 0, stream>>>(trans, labels, feats, chunkP, scores);
    reduce_mean_kernel<<<1, 32, 0, stream>>>(scores, (float*)d_out);
}